// SimpleRoutedExperts_25194278158789
// MI455X (gfx1250) — compile-verified
//
#include <hip/hip_runtime.h>
#include <hip/hip_bf16.h>

typedef __attribute__((ext_vector_type(2))) float v2f;
typedef __attribute__((ext_vector_type(8))) float v8f;

#define TT 4096
#define DD 2048
#define HH 1408
#define EE 8
#define KK 2

// ---- CDNA5 async global->LDS copy (16B per lane), tracked by ASYNCcnt ----
__device__ __forceinline__ void async_cp16(uint32_t lds_byte, const float* gbase,
                                           uint32_t gbyte_off) {
    asm volatile("global_load_async_to_lds_b128 %0, %1, %2"
                 :: "v"(lds_byte), "v"(gbyte_off), "s"(gbase)
                 : "memory");
}
#define WAIT_ASYNC(n) asm volatile("s_wait_asynccnt %0" :: "i"(n) : "memory")

#define WMMA_F32(A, B, C) \
    __builtin_amdgcn_wmma_f32_16x16x4_f32(false, (A), false, (B), (short)0, (C), false, false)

// ---------------- routing: build per-expert token lists ----------------
__global__ void moe_zero_counts(int* counts) {
    if (threadIdx.x < EE) counts[threadIdx.x] = 0;
}

__global__ void moe_build_lists(const int* __restrict__ indices,
                                const float* __restrict__ weights,
                                int* __restrict__ counts,
                                int* __restrict__ tokL,
                                float* __restrict__ gwL) {
    int i = blockIdx.x * 256 + threadIdx.x;      // over T*K pairs
    if (i >= TT * KK) return;
    int t = i >> 1;
    int e = indices[i];
    int pos = atomicAdd(&counts[e], 1);
    tokL[e * TT + pos] = t;
    gwL[e * TT + pos] = weights[i];
}

// ---------------- fused expert MLP ----------------
// block = 16 routed rows of one expert, 256 threads = 8 waves (wave32).
// Each wave streams its own 32-row weight slab with double-buffered async
// global->LDS copies; inner K loops are barrier-free.
__global__ __launch_bounds__(256)
void moe_expert_mlp(const float* __restrict__ x,
                    const float* __restrict__ w1,
                    const float* __restrict__ w2,
                    const int* __restrict__ counts,
                    const int* __restrict__ tokL,
                    const float* __restrict__ gwL,
                    float* __restrict__ y) {
    __shared__ float xs[16 * 2052];           // 131328 B  (x slab, padded stride)
    __shared__ float wtile[8 * 2 * 32 * 36];  //  73728 B  (per-wave double buffers)
    __shared__ float acts[16 * 1412];         //  90368 B  (gated activation)
    __shared__ int   stok[16];
    __shared__ float sgw[16];

    const int e = blockIdx.y;
    const int cnt = counts[e];
    const int n0 = blockIdx.x * 16;
    if (n0 >= cnt) return;
    const int rows = min(16, cnt - n0);

    const int tid = threadIdx.x;
    const int w = tid >> 5;
    const int lane = tid & 31;
    const int nloc = lane & 15;
    const int khalf = lane >> 4;

    if (tid < 16) {
        int t = 0; float g = 0.0f;
        if (tid < rows) { t = tokL[e * TT + n0 + tid]; g = gwL[e * TT + n0 + tid]; }
        stok[tid] = t;
        sgw[tid] = g;
    }
    __syncthreads();

    // ---- stage the 16 x rows (full K=2048) into LDS once, zero-pad tail rows ----
    #pragma unroll
    for (int j = 0; j < 32; ++j) {                 // 16 rows * 512 float4
        int flat = j * 256 + tid;
        int r = flat >> 9;
        int c4 = flat & 511;
        float4 v = make_float4(0.f, 0.f, 0.f, 0.f);
        if (r < rows)
            v = ((const float4*)(x + (size_t)stok[r] * DD))[c4];
        *(float4*)&xs[r * 2052 + c4 * 4] = v;
    }
    __syncthreads();

    const float* w1e = w1 + (size_t)e * (2 * HH) * DD;
    const float* w2e = w2 + (size_t)e * DD * HH;

    // per-wave LDS weight-buffer bases
    const uint32_t wtb = (uint32_t)(uintptr_t)&wtile[0] + (uint32_t)w * 9216u; // bytes
    const int wtf = w * 2304;                                                  // floats
    const uint32_t c4b = (uint32_t)(lane & 7) * 16u;   // byte column within slab
    const int rb = lane >> 3;                          // base row for staging

    // ---- phase 1: h = x @ w1^T, gated act = u * silu(g) * combine_weight ----
    // wave w owns u-tile (p*8+w) and matching g-tile (+88); 64 K-slabs of 32.
    for (int p = 0; p < 11; ++p) {
        v8f accu = {};
        v8f accg = {};

        // stage(buf, kc): 32 rows (16 u + 16 g) x 32 cols, 8 async b128 per lane
        auto stage1 = [&](uint32_t bufbyte, int kc) {
            #pragma unroll
            for (int j = 0; j < 8; ++j) {
                int rr = rb + 4 * j;                       // 0..31
                int srcn = (rr < 16) ? (p * 128 + w * 16 + rr)
                                     : (HH + p * 128 + w * 16 + (rr - 16));
                uint32_t goff = (uint32_t)(srcn * DD + kc) * 4u + c4b;
                async_cp16(bufbyte + (uint32_t)rr * 144u + c4b, w1e, goff);
            }
        };

        stage1(wtb, 0);
        for (int it = 0; it < 63; ++it) {
            int cur = it & 1;
            stage1(wtb + (uint32_t)(cur ^ 1) * 4608u, (it + 1) * 32);
            WAIT_ASYNC(8);                 // in-order: current slab has landed
            const int cb = wtf + cur * 1152;
            const int ac = it * 32;
            #pragma unroll
            for (int kk = 0; kk < 32; kk += 4) {
                v2f a  = *(const v2f*)&xs[nloc * 2052 + ac + kk + khalf * 2];
                v2f bu = *(const v2f*)&wtile[cb + nloc * 36 + kk + khalf * 2];
                v2f bg = *(const v2f*)&wtile[cb + 576 + nloc * 36 + kk + khalf * 2];
                accu = WMMA_F32(a, bu, accu);
                accg = WMMA_F32(a, bg, accg);
            }
        }
        WAIT_ASYNC(0);
        {   // tail slab: it = 63, buffer 1, kc = 2016
            const int cb = wtf + 1152;
            #pragma unroll
            for (int kk = 0; kk < 32; kk += 4) {
                v2f a  = *(const v2f*)&xs[nloc * 2052 + 2016 + kk + khalf * 2];
                v2f bu = *(const v2f*)&wtile[cb + nloc * 36 + kk + khalf * 2];
                v2f bg = *(const v2f*)&wtile[cb + 576 + nloc * 36 + kk + khalf * 2];
                accu = WMMA_F32(a, bu, accu);
                accg = WMMA_F32(a, bg, accg);
            }
        }

        // u * silu(g) * cw -> act tile
        #pragma unroll
        for (int v = 0; v < 8; ++v) {
            int m = khalf * 8 + v;
            float uu = accu[v];
            float gg = accg[v];
            float sg = gg / (1.0f + expf(-gg));
            acts[m * 1412 + (p * 8 + w) * 16 + nloc] = uu * sg * sgw[m];
        }
    }

    __syncthreads();   // all act tiles visible to all waves

    // ---- phase 2: out = act @ w2^T, atomic-accumulate into y ----
    for (int q = 0; q < 16; ++q) {
        v8f acc = {};

        auto stage2 = [&](uint32_t bufbyte, int kc) {   // 16 rows x 32 cols, 4 async/lane
            #pragma unroll
            for (int j = 0; j < 4; ++j) {
                int rr = rb + 4 * j;                    // 0..15
                int srcn = q * 128 + w * 16 + rr;
                uint32_t goff = (uint32_t)(srcn * HH + kc) * 4u + c4b;
                async_cp16(bufbyte + (uint32_t)rr * 144u + c4b, w2e, goff);
            }
        };

        stage2(wtb, 0);
        for (int it = 0; it < 43; ++it) {               // 44 K-slabs of 32
            int cur = it & 1;
            stage2(wtb + (uint32_t)(cur ^ 1) * 4608u, (it + 1) * 32);
            WAIT_ASYNC(4);
            const int cb = wtf + cur * 1152;
            const int ac = it * 32;
            #pragma unroll
            for (int kk = 0; kk < 32; kk += 4) {
                v2f a = *(const v2f*)&acts[nloc * 1412 + ac + kk + khalf * 2];
                v2f b = *(const v2f*)&wtile[cb + nloc * 36 + kk + khalf * 2];
                acc = WMMA_F32(a, b, acc);
            }
        }
        WAIT_ASYNC(0);
        {   // tail slab: it = 43, buffer 1, kc = 1376
            const int cb = wtf + 1152;
            #pragma unroll
            for (int kk = 0; kk < 32; kk += 4) {
                v2f a = *(const v2f*)&acts[nloc * 1412 + 1376 + kk + khalf * 2];
                v2f b = *(const v2f*)&wtile[cb + nloc * 36 + kk + khalf * 2];
                acc = WMMA_F32(a, b, acc);
            }
        }

        #pragma unroll
        for (int v = 0; v < 8; ++v) {
            int m = khalf * 8 + v;
            if (m < rows) {
                int col = q * 128 + w * 16 + nloc;
                atomicAdd(&y[(size_t)stok[m] * DD + col], acc[v]);
            }
        }
    }
}

extern "C" void kernel_launch(void* const* d_in, const int* in_sizes, int n_in,
                              void* d_out, int out_size, void* d_ws, size_t ws_size,
                              hipStream_t stream) {
    const float* x       = (const float*)d_in[0];
    const float* weights = (const float*)d_in[1];
    const int*   indices = (const int*)d_in[2];
    const float* w1      = (const float*)d_in[3];
    const float* w2      = (const float*)d_in[4];
    float* y = (float*)d_out;

    char* ws = (char*)d_ws;
    int*   cntp = (int*)ws;                              // 8 ints (256 B slot)
    int*   tokL = (int*)(ws + 256);                      // E*T ints
    float* gwL  = (float*)(ws + 256 + EE * TT * 4);      // E*T floats

    hipMemsetAsync(d_out, 0, (size_t)out_size * sizeof(float), stream);

    moe_zero_counts<<<1, 32, 0, stream>>>(cntp);
    moe_build_lists<<<(TT * KK + 255) / 256, 256, 0, stream>>>(
        indices, weights, cntp, tokL, gwL);

    dim3 grid(TT / 16, EE);
    moe_expert_mlp<<<grid, 256, 0, stream>>>(x, w1, w2, cntp, tokL, gwL, y);
}